// SASRec_CoLR_23124103921735
// MI455X (gfx1250) — compile-verified
//
#include <hip/hip_runtime.h>
#include <math.h>

#define BATCH 128
#define SEQL  200
#define LP    208   // padded seq length (13*16)
#define HID   64
#define RANK  8
#define NTM   13    // LP/16 row tiles
#define NTN   4     // HID/16 col tiles

typedef __attribute__((ext_vector_type(2))) float v2f;
typedef __attribute__((ext_vector_type(8))) float v8f;

__device__ __forceinline__ v8f wmma_f32(v2f a, v2f b, v8f c) {
  // V_WMMA_F32_16X16X4_F32: D(16x16) = A(16x4) * B(4x16) + C
  return __builtin_amdgcn_wmma_f32_16x16x4_f32(
      false, a, false, b, (short)0, c, false, false);
}

__device__ __forceinline__ float wave_sum32(float v) {
#pragma unroll
  for (int o = 16; o >= 1; o >>= 1) v += __shfl_xor(v, o, 32);
  return v;
}

// ---------------------------------------------------------------------------
// Build seqs: X[b][l][h] = (A[idx]·B[:,h] + item_emb[idx][h]) * 8 + pos_emb[p][h]
// Pad rows (l >= 200) set to zero. One wave per row, 2 h-values per lane.
// ---------------------------------------------------------------------------
__global__ __launch_bounds__(256)
void embed_kernel(const int* __restrict__ log_seqs,
                  const float* __restrict__ item_emb,
                  const float* __restrict__ pos_emb,
                  const float* __restrict__ Am,
                  const float* __restrict__ Bm,
                  float* __restrict__ X) {
  int row  = blockIdx.x * 8 + (threadIdx.x >> 5);   // 0..BATCH*LP-1
  int b    = row / LP;
  int l    = row - b * LP;
  int lane = threadIdx.x & 31;
  float* xr = X + (size_t)row * HID;
  if (l >= SEQL) { xr[lane] = 0.f; xr[lane + 32] = 0.f; return; }
  int idx = log_seqs[b * SEQL + l];
  int p   = (idx != 0) ? (l + 1) : 0;
  float e0 = item_emb[(size_t)idx * HID + lane];
  float e1 = item_emb[(size_t)idx * HID + lane + 32];
#pragma unroll
  for (int r = 0; r < RANK; ++r) {
    float ar = Am[(size_t)idx * RANK + r];
    e0 += ar * Bm[r * HID + lane];
    e1 += ar * Bm[r * HID + lane + 32];
  }
  xr[lane]      = e0 * 8.0f + pos_emb[(size_t)p * HID + lane];
  xr[lane + 32] = e1 * 8.0f + pos_emb[(size_t)p * HID + lane + 32];
}

// ---------------------------------------------------------------------------
// LayerNorm over 64 features. One wave per row (covers pad rows: finite).
// ---------------------------------------------------------------------------
__global__ __launch_bounds__(256)
void ln_kernel(const float* __restrict__ X,
               const float* __restrict__ g,
               const float* __restrict__ bt,
               float* __restrict__ Y) {
  int row  = blockIdx.x * 8 + (threadIdx.x >> 5);
  int lane = threadIdx.x & 31;
  const float* x = X + (size_t)row * HID;
  float x0 = x[lane], x1 = x[lane + 32];
  float mu = wave_sum32(x0 + x1) * (1.0f / 64.0f);
  float d0 = x0 - mu, d1 = x1 - mu;
  float var = wave_sum32(d0 * d0 + d1 * d1) * (1.0f / 64.0f);
  float inv = rsqrtf(var + 1e-8f);
  float* y = Y + (size_t)row * HID;
  y[lane]      = d0 * inv * g[lane] + bt[lane];
  y[lane + 32] = d1 * inv * g[lane + 32] + bt[lane + 32];
}

// ---------------------------------------------------------------------------
// Y = X @ W^T + bias [, relu] [, + res]. X,Y: [BATCH][LP][64], W: [64][64].
// One wave per 16x16 output tile; 16 WMMA f32 16x16x4 k-steps.
// ---------------------------------------------------------------------------
__global__ __launch_bounds__(128)
void gemm64_kernel(const float* __restrict__ X,
                   const float* __restrict__ W,
                   const float* __restrict__ bias,
                   const float* __restrict__ res,
                   float* __restrict__ Y, int relu) {
  int job  = blockIdx.x * 4 + (threadIdx.x >> 5);   // BATCH*NTM*NTN jobs
  int nt   = job & 3;
  int t    = job >> 2;
  int mt   = t % NTM;
  int b    = t / NTM;
  int lane = threadIdx.x & 31;
  int lh   = lane & 15;
  bool hi  = lane >= 16;

  const float* Xb = X + (size_t)b * LP * HID;
  int arow = mt * 16 + lh;
  int bcol = nt * 16 + lh;

  v8f acc = {};
#pragma unroll
  for (int k0 = 0; k0 < HID; k0 += 4) {
    int ka = k0 + (hi ? 2 : 0);
    v2f a  = *(const v2f*)(Xb + (size_t)arow * HID + ka);  // A[m][k..k+1]
    v2f bb = *(const v2f*)(W + (size_t)bcol * HID + ka);   // B[k][n] = W[n][k]
    acc = wmma_f32(a, bb, acc);
  }

  float bv = bias[bcol];
  float* Yb = Y + (size_t)b * LP * HID;
  const float* Rb = res ? res + (size_t)b * LP * HID : nullptr;
#pragma unroll
  for (int r = 0; r < 8; ++r) {
    int m = mt * 16 + r + (hi ? 8 : 0);
    float v = acc[r] + bv;
    if (relu) v = fmaxf(v, 0.0f);
    if (Rb) v += Rb[(size_t)m * HID + bcol];
    Yb[(size_t)m * HID + bcol] = v;
  }
}

// ---------------------------------------------------------------------------
// Causal attention (1 head, hd=64). One wave per (batch, 16-row q-tile):
//   S = Q_tile @ K^T * 0.125 (WMMA, all causal jtiles kept in registers),
//   mask, row softmax (half-wave shuffles), P -> LDS (D->A layout),
//   CTX_tile = P @ V (WMMA). CTX written in place over Q rows (safe: wave
//   only reads its own q rows, all before the store).
// ---------------------------------------------------------------------------
__global__ __launch_bounds__(32)
void attn_kernel(const float* __restrict__ Q,
                 const float* __restrict__ K,
                 const float* __restrict__ V,
                 float* __restrict__ CTX) {
  __shared__ float Psh[16 * LP];
  int mt   = blockIdx.x % NTM;
  int b    = blockIdx.x / NTM;
  int lane = threadIdx.x;
  int lh   = lane & 15;
  bool hi  = lane >= 16;
  int njt  = mt + 1;

  const float* Qb = Q + (size_t)b * LP * HID;
  const float* Kb = K + (size_t)b * LP * HID;
  const float* Vb = V + (size_t)b * LP * HID;

  v8f S[NTM];
#pragma unroll
  for (int jt = 0; jt < NTM; ++jt) {
    if (jt < njt) {
      v8f acc = {};
      for (int k0 = 0; k0 < HID; k0 += 4) {
        int ka = k0 + (hi ? 2 : 0);
        v2f a  = *(const v2f*)(Qb + (size_t)(mt * 16 + lh) * HID + ka);
        v2f bb = *(const v2f*)(Kb + (size_t)(jt * 16 + lh) * HID + ka);
        acc = wmma_f32(a, bb, acc);
      }
      S[jt] = acc;
    }
  }

  // scale + causal mask (also mask pad keys j >= 200)
#pragma unroll
  for (int jt = 0; jt < NTM; ++jt) {
    if (jt < njt) {
      int j = jt * 16 + lh;
#pragma unroll
      for (int r = 0; r < 8; ++r) {
        int m = mt * 16 + r + (hi ? 8 : 0);
        float v = S[jt][r] * 0.125f;
        S[jt][r] = (j <= m && j < SEQL) ? v : -3.0e38f;
      }
    }
  }

  // row softmax: row m's 16*njt values live in one 16-lane half
#pragma unroll
  for (int r = 0; r < 8; ++r) {
    float mx = -3.0e38f;
#pragma unroll
    for (int jt = 0; jt < NTM; ++jt)
      if (jt < njt) mx = fmaxf(mx, S[jt][r]);
#pragma unroll
    for (int o = 8; o >= 1; o >>= 1) mx = fmaxf(mx, __shfl_xor(mx, o, 32));
    float sum = 0.f;
#pragma unroll
    for (int jt = 0; jt < NTM; ++jt) {
      if (jt < njt) {
        float e = __expf(S[jt][r] - mx);
        S[jt][r] = e;
        sum += e;
      }
    }
#pragma unroll
    for (int o = 8; o >= 1; o >>= 1) sum += __shfl_xor(sum, o, 32);
    float inv = 1.0f / sum;
    int mrow = r + (hi ? 8 : 0);
#pragma unroll
    for (int jt = 0; jt < NTM; ++jt)
      if (jt < njt) Psh[mrow * LP + jt * 16 + lh] = S[jt][r] * inv;
  }
  __syncthreads();

  // CTX_tile = P @ V
  int jmax = njt * 16;
#pragma unroll
  for (int nt = 0; nt < NTN; ++nt) {
    v8f acc = {};
    int n = nt * 16 + lh;
    for (int j0 = 0; j0 < jmax; j0 += 4) {
      int ja = j0 + (hi ? 2 : 0);
      v2f a = *(const v2f*)&Psh[lh * LP + ja];          // A[m][j..j+1]
      v2f bb;
      bb.x = Vb[(size_t)ja * HID + n];                  // B[j][n] = V[j][n]
      bb.y = Vb[(size_t)(ja + 1) * HID + n];
      acc = wmma_f32(a, bb, acc);
    }
    float* Cb = CTX + (size_t)b * LP * HID;
#pragma unroll
    for (int r = 0; r < 8; ++r) {
      int m = mt * 16 + r + (hi ? 8 : 0);
      Cb[(size_t)m * HID + n] = acc[r];
    }
  }
}

// ---------------------------------------------------------------------------
// logits[b][l] = feats[b][l] . (A[idx]@B + item_emb[idx]) for pos & neg.
// One wave per (b,l); output: [pos 25600 | neg 25600].
// ---------------------------------------------------------------------------
__global__ __launch_bounds__(256)
void logits_kernel(const int* __restrict__ pos_seqs,
                   const int* __restrict__ neg_seqs,
                   const float* __restrict__ feats,
                   const float* __restrict__ item_emb,
                   const float* __restrict__ Am,
                   const float* __restrict__ Bm,
                   float* __restrict__ out) {
  int row  = blockIdx.x * 8 + (threadIdx.x >> 5);   // 0..BATCH*SEQL-1
  int b    = row / SEQL;
  int l    = row - b * SEQL;
  int lane = threadIdx.x & 31;
  const float* f = feats + ((size_t)b * LP + l) * HID;
  float f0 = f[lane], f1 = f[lane + 32];
#pragma unroll
  for (int which = 0; which < 2; ++which) {
    int idx = (which ? neg_seqs : pos_seqs)[row];
    float e0 = item_emb[(size_t)idx * HID + lane];
    float e1 = item_emb[(size_t)idx * HID + lane + 32];
#pragma unroll
    for (int r = 0; r < RANK; ++r) {
      float ar = Am[(size_t)idx * RANK + r];
      e0 += ar * Bm[r * HID + lane];
      e1 += ar * Bm[r * HID + lane + 32];
    }
    float s = wave_sum32(f0 * e0 + f1 * e1);
    if (lane == 0) out[which * (BATCH * SEQL) + row] = s;
  }
}

// ---------------------------------------------------------------------------
extern "C" void kernel_launch(void* const* d_in, const int* in_sizes, int n_in,
                              void* d_out, int out_size, void* d_ws, size_t ws_size,
                              hipStream_t stream) {
  (void)in_sizes; (void)n_in; (void)out_size; (void)ws_size;
  const int*   log_seqs  = (const int*)d_in[1];
  const int*   pos_seqs  = (const int*)d_in[2];
  const int*   neg_seqs  = (const int*)d_in[3];
  const float* item_emb  = (const float*)d_in[4];
  const float* pos_emb   = (const float*)d_in[5];
  const float* Amat      = (const float*)d_in[6];
  const float* Bmat      = (const float*)d_in[7];
  const float* attn_ln_g = (const float*)d_in[8];
  const float* attn_ln_b = (const float*)d_in[9];
  const float* in_proj_w = (const float*)d_in[10];
  const float* in_proj_b = (const float*)d_in[11];
  const float* out_w     = (const float*)d_in[12];
  const float* out_b     = (const float*)d_in[13];
  const float* fwd_ln_g  = (const float*)d_in[14];
  const float* fwd_ln_b  = (const float*)d_in[15];
  const float* c1_w      = (const float*)d_in[16];
  const float* c1_b      = (const float*)d_in[17];
  const float* c2_w      = (const float*)d_in[18];
  const float* c2_b      = (const float*)d_in[19];
  const float* last_g    = (const float*)d_in[20];
  const float* last_b    = (const float*)d_in[21];
  float* out = (float*)d_out;

  size_t bufE = (size_t)BATCH * LP * HID;    // 1,703,936 floats per buffer
  float* ws = (float*)d_ws;
  float* X  = ws;                 // current seqs
  float* XL = ws + 1 * bufE;      // LN output (also residual source)
  float* Qb = ws + 2 * bufE;      // q / ctx / FFN hidden
  float* Kb = ws + 3 * bufE;
  float* Vb = ws + 4 * bufE;

  const int rowBlocks  = BATCH * LP / 8;        // 3328 (256-thread blocks, 8 rows each)
  const int gemmBlocks = BATCH * NTM * NTN / 4; // 1664 (128-thread blocks, 4 tiles each)

  embed_kernel<<<rowBlocks, 256, 0, stream>>>(log_seqs, item_emb, pos_emb, Amat, Bmat, X);

  for (int i = 0; i < 2; ++i) {
    ln_kernel<<<rowBlocks, 256, 0, stream>>>(X, attn_ln_g + i * HID, attn_ln_b + i * HID, XL);
    const float* wq = in_proj_w + (size_t)i * 3 * HID * HID;
    const float* bq = in_proj_b + (size_t)i * 3 * HID;
    gemm64_kernel<<<gemmBlocks, 128, 0, stream>>>(XL, wq,               bq,           nullptr, Qb, 0);
    gemm64_kernel<<<gemmBlocks, 128, 0, stream>>>(X,  wq + HID * HID,   bq + HID,     nullptr, Kb, 0);
    gemm64_kernel<<<gemmBlocks, 128, 0, stream>>>(X,  wq + 2 * HID * HID, bq + 2 * HID, nullptr, Vb, 0);
    attn_kernel<<<BATCH * NTM, 32, 0, stream>>>(Qb, Kb, Vb, Qb);
    gemm64_kernel<<<gemmBlocks, 128, 0, stream>>>(Qb, out_w + (size_t)i * HID * HID, out_b + i * HID, XL, X, 0);
    ln_kernel<<<rowBlocks, 256, 0, stream>>>(X, fwd_ln_g + i * HID, fwd_ln_b + i * HID, XL);
    gemm64_kernel<<<gemmBlocks, 128, 0, stream>>>(XL, c1_w + (size_t)i * HID * HID, c1_b + i * HID, nullptr, Qb, 1);
    gemm64_kernel<<<gemmBlocks, 128, 0, stream>>>(Qb, c2_w + (size_t)i * HID * HID, c2_b + i * HID, XL, X, 0);
  }

  ln_kernel<<<rowBlocks, 256, 0, stream>>>(X, last_g, last_b, XL);
  logits_kernel<<<BATCH * SEQL / 8, 256, 0, stream>>>(pos_seqs, neg_seqs, XL, item_emb, Amat, Bmat, out);
}